// SimplicialCN_25821343384027
// MI455X (gfx1250) — compile-verified
//
#include <hip/hip_runtime.h>
#include <hip/hip_bf16.h>

// GCN2Conv fused pipeline for MI455X (gfx1250, wave32)
//   h   = alpha*x0  (init)
//   h  += (1-alpha) * edge_weight[e] * x[src[e]]  scattered to dst[e]   (atomics)
//   out = (1-beta)*h + beta*(h @ W1)              (fp32 WMMA 16x16x4 GEMM)

#define CH        512
#define ALPHA_F   0.1f
#define BETA_F    0.4054651081081644f          // log(1.5)
#define OMB_F     0.5945348918918356f          // 1 - beta

typedef __attribute__((ext_vector_type(2))) float v2f;
typedef __attribute__((ext_vector_type(8))) float v8f;

// ---------------------------------------------------------------- init: h = alpha*x0
__global__ __launch_bounds__(256) void gcn2_init(const float* __restrict__ x0,
                                                 float* __restrict__ hbuf, int n4) {
    int i = blockIdx.x * 256 + threadIdx.x;
    if (i < n4) {
        float4 v = ((const float4*)x0)[i];
        float4 r;
        r.x = ALPHA_F * v.x; r.y = ALPHA_F * v.y;
        r.z = ALPHA_F * v.z; r.w = ALPHA_F * v.w;
        ((float4*)hbuf)[i] = r;
    }
}

// ------------------------------------------------- scatter: h[dst] += (1-a)*w * x[src]
// One wave32 per edge; lanes sweep 512 channels as float4 (coalesced 512B bursts),
// scalar global_atomic_add_f32 into the destination row (resolved in L2).
__global__ __launch_bounds__(256) void gcn2_scatter(const float* __restrict__ x,
                                                    const float* __restrict__ ew,
                                                    const int*   __restrict__ ei,
                                                    float* __restrict__ hbuf, int E) {
    int e = blockIdx.x * 8 + (threadIdx.x >> 5);
    if (e >= E) return;
    int lane = threadIdx.x & 31;
    int src = ei[e];
    int dst = ei[E + e];
    float s = (1.0f - ALPHA_F) * ew[e];
    const float4* xs = (const float4*)(x + (size_t)src * CH);
    float*        hd = hbuf + (size_t)dst * CH;
#pragma unroll
    for (int c = 0; c < CH / 4; c += 32) {
        float4 v = xs[c + lane];
        int base = (c + lane) * 4;
        atomicAdd(hd + base + 0, v.x * s);
        atomicAdd(hd + base + 1, v.y * s);
        atomicAdd(hd + base + 2, v.z * s);
        atomicAdd(hd + base + 3, v.w * s);
    }
}

// --------------------------------------- GEMM: out = (1-beta)*h + beta*(h @ W1)
// Block = 256 threads = 8 waves. Block tile: 128(M) x 128(N).
// Wave w owns rows [m0 + 16w, m0 + 16w + 16), 8 n-tiles of 16 => 8 v8f accumulators.
// K staged through LDS in chunks of 32: As[128][36] row-major, Bs transposed [128n][36k]
// so a lane's B fragment (B[k][n], B[k+1][n]) is one aligned float2 read.
#define KT   32
#define LDP  36   // padded LDS stride (floats); row byte stride 144 = 16B aligned

__global__ __launch_bounds__(256) void gcn2_gemm(const float* __restrict__ h,
                                                 const float* __restrict__ W1,
                                                 float* __restrict__ out, int M) {
    __shared__ float As[128 * LDP];
    __shared__ float Bs[128 * LDP];

    const int tid  = threadIdx.x;
    const int wave = tid >> 5;
    const int lane = tid & 31;
    const int l15  = lane & 15;
    const int hi   = lane >> 4;          // 0: lanes 0-15, 1: lanes 16-31
    const int m0   = blockIdx.x * 128;
    const int n0   = blockIdx.y * 128;
    const int mrow = m0 + wave * 16;     // wave-uniform

    v8f acc[8] = {};

    for (int kb = 0; kb < CH; kb += KT) {
        // --- stage A panel: h[m0:m0+128, kb:kb+32]  (float4, fully coalesced)
#pragma unroll
        for (int i = 0; i < 4; ++i) {
            int r  = (tid >> 3) + i * 32;        // 0..127
            int f4 = tid & 7;                    // 0..7 -> k offset f4*4
            int gm = m0 + r;
            float4 v = make_float4(0.f, 0.f, 0.f, 0.f);
            if (gm < M) v = *(const float4*)(h + (size_t)gm * CH + kb + f4 * 4);
            *(float4*)(&As[r * LDP + f4 * 4]) = v;
        }
        // --- stage B panel transposed: Bs[n][k] = W1[kb+k][n0+n]
#pragma unroll
        for (int i = 0; i < 16; ++i) {
            int linear = tid + i * 256;          // 0..4095
            int n = linear & 127;
            int k = linear >> 7;                 // 0..31
            Bs[n * LDP + k] = W1[(size_t)(kb + k) * CH + n0 + n];
        }
        __syncthreads();

        if (mrow < M) {                          // wave-uniform: EXEC stays all-1s
#pragma unroll
            for (int kk = 0; kk < KT; kk += 4) {
                int kOff = kk + (hi << 1);       // ISA 16x4 f32 fragment layout
                v2f a = *(const v2f*)(&As[(wave * 16 + l15) * LDP + kOff]);
#pragma unroll
                for (int nt = 0; nt < 8; ++nt) {
                    v2f b = *(const v2f*)(&Bs[(nt * 16 + l15) * LDP + kOff]);
                    acc[nt] = __builtin_amdgcn_wmma_f32_16x16x4_f32(
                        false, a, false, b, (short)0, acc[nt], false, false);
                }
            }
        }
        __syncthreads();
    }

    // --- fused epilogue: out = (1-beta)*h + beta*acc
    if (mrow < M) {
#pragma unroll
        for (int nt = 0; nt < 8; ++nt) {
#pragma unroll
            for (int v = 0; v < 8; ++v) {
                int m = mrow + v + (hi << 3);    // C/D layout: vgpr v, half-wave -> +8
                int n = n0 + nt * 16 + l15;
                size_t idx = (size_t)m * CH + n;
                out[idx] = OMB_F * h[idx] + BETA_F * acc[nt][v];
            }
        }
    }
}

// ---------------------------------------------------------------------------- launch
extern "C" void kernel_launch(void* const* d_in, const int* in_sizes, int n_in,
                              void* d_out, int out_size, void* d_ws, size_t ws_size,
                              hipStream_t stream) {
    const float* x   = (const float*)d_in[0];
    const float* x0  = (const float*)d_in[1];
    const float* W1  = (const float*)d_in[2];
    const float* ew  = (const float*)d_in[3];
    const int*   ei  = (const int*)  d_in[4];
    float*       out = (float*)d_out;
    float*       hbf = (float*)d_ws;             // N*CH floats of scratch

    const int N = in_sizes[0] / CH;              // 50000
    const int E = in_sizes[3];                   // 800000

    // h = alpha * x0
    int n4 = (N * CH) / 4;
    gcn2_init<<<(n4 + 255) / 256, 256, 0, stream>>>(x0, hbf, n4);

    // h[dst] += (1-alpha)*w * x[src]   (8 edges per block, one wave each)
    gcn2_scatter<<<(E + 7) / 8, 256, 0, stream>>>(x, ew, ei, hbf, E);

    // out = (1-beta)*h + beta*(h @ W1)
    dim3 grid((N + 127) / 128, CH / 128);
    gcn2_gemm<<<grid, 256, 0, stream>>>(hbf, W1, out, N);
}